// SRUPlusPlus_86320252715772
// MI455X (gfx1250) — compile-verified
//
#include <hip/hip_runtime.h>
#include <hip/hip_bf16.h>

typedef float v2f __attribute__((ext_vector_type(2)));
typedef float v8f __attribute__((ext_vector_type(8)));

// ---------------------------------------------------------------------------
// WMMA f32 GEMM:  C[M,N] = A[M,K] * op(B)  (+ bias[N] if bias != nullptr)
//   BT=true : B stored [N,K] row-major, op(B)=B^T   (x @ W.T, Q @ K^T)
//   BT=false: B stored [K,N] row-major, op(B)=B     (attn @ V)
//
// Block = 256 threads = 8 waves -> 128(M) x 64(N) tile; wave w owns 16 rows.
// K is consumed in slabs of 16 (4 x V_WMMA_F32_16X16X4_F32 per fragment set).
// B slab (64x16) is staged in LDS with GLOBAL_LOAD_ASYNC_TO_LDS_B128, double
// buffered: the async copy of slab ks+1 overlaps the 16 WMMAs of slab ks.
// A fragments are register-prefetched one slab ahead (plain b64 loads).
//
// Fragment layout per CDNA5 ISA 7.12.2 (wave32):
//   A (16x4): lane holds A[lane&15][k + 2*(lane>>4) + {0,1}]      -> v2f
//   B (4x16): lane holds B[k + 2*(lane>>4) + {0,1}][n0 + lane&15] -> v2f
//   C (16x16): vgpr r, lane -> C[r + 8*(lane>>4)][lane&15]
// ---------------------------------------------------------------------------
template <bool BT>
__global__ __launch_bounds__(256) void
sru_gemm_wmma(const float* __restrict__ A, int lda, long sA,
              const float* __restrict__ Bm, int ldb, long sB,
              float* __restrict__ C, int ldc, long sC,
              const float* __restrict__ bias,
              int M, int N, int K)
{
    // LDS B-slab layouts (double buffered):
    //   BT: [n=0..63][k=0..15], row stride 20 floats (80B: b128-store aligned,
    //       conflict-free b64 fragment reads)
    //   NN: [k=0..15][n=0..63], row stride 72 floats (288B: aligned, and the
    //       two half-waves hit disjoint bank groups on fragment reads)
    constexpr int LSTR = BT ? 20 : 72;
    constexpr int LROW = BT ? 64 : 16;
    __shared__ float sBuf[2][LROW * LSTR];

    const int batch = blockIdx.y;
    A  += (long)batch * sA;
    Bm += (long)batch * sB;
    C  += (long)batch * sC;

    const int t    = threadIdx.x;
    const int lane = t & 31;
    const int wave = t >> 5;
    const int gm   = M >> 7;                    // 128-row block tiles
    const int mi   = blockIdx.x % gm;
    const int ni   = blockIdx.x / gm;
    const int m0   = (mi << 7) + (wave << 4);   // this wave's 16 rows
    const int n0   = ni << 6;

    const int half = lane >> 4;                 // 0: lanes 0-15, 1: 16-31
    const int l16  = lane & 15;
    const int koff = half << 1;                 // K sub-offset 0 or 2

    // --- global->LDS async copy mapping: one b128 per thread per slab ------
    const float* gB;
    unsigned ldsIdx;
    long gStep;
    if constexpr (BT) {
        const int bn = t >> 2, bk = (t & 3) << 2;       // B row n, k-chunk
        gB     = Bm + (long)(n0 + bn) * ldb + bk;
        ldsIdx = (unsigned)(bn * LSTR + bk);
        gStep  = 16;                                    // k advances by 16
    } else {
        const int bk = t >> 4, bn = (t & 15) << 2;      // B row k, n-chunk
        gB     = Bm + (long)bk * ldb + (n0 + bn);
        ldsIdx = (unsigned)(bk * LSTR + bn);
        gStep  = (long)16 * ldb;                        // 16 k-rows
    }
    // LDS byte offsets for the two buffers (low 32 bits of the generic
    // address of a __shared__ object are the LDS offset on gfx1250).
    const unsigned ldsOff0 = (unsigned)(uintptr_t)(&sBuf[0][ldsIdx]);
    const unsigned ldsOff1 = (unsigned)(uintptr_t)(&sBuf[1][ldsIdx]);

    v8f acc0 = {}, acc1 = {}, acc2 = {}, acc3 = {};
    const float* arow = A + (long)(m0 + l16) * lda + koff;
    const int nslab = K >> 4;

    // --- prologue: slab 0 -> buf0, A fragments for slab 0 ------------------
    asm volatile("global_load_async_to_lds_b128 %0, %1, off"
                 :: "v"(ldsOff0), "v"(gB) : "memory");
    v2f a_cur0 = *(const v2f*)(arow + 0);
    v2f a_cur1 = *(const v2f*)(arow + 4);
    v2f a_cur2 = *(const v2f*)(arow + 8);
    v2f a_cur3 = *(const v2f*)(arow + 12);
    asm volatile("s_wait_asynccnt 0" ::: "memory");
    __syncthreads();

    for (int ks = 0; ks < nslab; ++ks) {
        const int P = ks & 1;
        const float* lbuf = sBuf[P];
        const bool more = (ks + 1) < nslab;

        // kick off slab ks+1: async B -> buf 1-P, A fragments -> regs
        v2f a_nxt0, a_nxt1, a_nxt2, a_nxt3;
        if (more) {
            const float* g = gB + (long)(ks + 1) * gStep;
            asm volatile("global_load_async_to_lds_b128 %0, %1, off"
                         :: "v"(P ? ldsOff0 : ldsOff1), "v"(g) : "memory");
            const float* ar = arow + (long)(ks + 1) * 16;
            a_nxt0 = *(const v2f*)(ar + 0);
            a_nxt1 = *(const v2f*)(ar + 4);
            a_nxt2 = *(const v2f*)(ar + 8);
            a_nxt3 = *(const v2f*)(ar + 12);
        }

        // B fragments for all 4 k-steps x 4 n-subtiles from LDS
        v2f bf[4][4];
#pragma unroll
        for (int kk = 0; kk < 4; ++kk) {
#pragma unroll
            for (int nt = 0; nt < 4; ++nt) {
                if constexpr (BT) {
                    bf[kk][nt] = *(const v2f*)&lbuf[(nt * 16 + l16) * LSTR +
                                                    kk * 4 + koff];
                } else {
                    v2f f;
                    f.x = lbuf[(kk * 4 + koff)     * LSTR + nt * 16 + l16];
                    f.y = lbuf[(kk * 4 + koff + 1) * LSTR + nt * 16 + l16];
                    bf[kk][nt] = f;
                }
            }
        }

        // 16 back-to-back WMMAs
        v2f ak;
#pragma unroll
        for (int kk = 0; kk < 4; ++kk) {
            ak = (kk == 0) ? a_cur0 : (kk == 1) ? a_cur1 : (kk == 2) ? a_cur2 : a_cur3;
            acc0 = __builtin_amdgcn_wmma_f32_16x16x4_f32(false, ak, false, bf[kk][0], (short)0, acc0, false, false);
            acc1 = __builtin_amdgcn_wmma_f32_16x16x4_f32(false, ak, false, bf[kk][1], (short)0, acc1, false, false);
            acc2 = __builtin_amdgcn_wmma_f32_16x16x4_f32(false, ak, false, bf[kk][2], (short)0, acc2, false, false);
            acc3 = __builtin_amdgcn_wmma_f32_16x16x4_f32(false, ak, false, bf[kk][3], (short)0, acc3, false, false);
        }

        // publish buf 1-P: my async copy landed, everyone past reads of buf P
        asm volatile("s_wait_asynccnt 0" ::: "memory");
        __syncthreads();

        if (more) { a_cur0 = a_nxt0; a_cur1 = a_nxt1; a_cur2 = a_nxt2; a_cur3 = a_nxt3; }
    }

    // --- epilogue: C tile store (+ optional bias) --------------------------
    const int row0 = m0 + (half << 3);
    const int col  = n0 + l16;
    float bv0 = 0.f, bv1 = 0.f, bv2 = 0.f, bv3 = 0.f;
    if (bias) { bv0 = bias[col]; bv1 = bias[col + 16]; bv2 = bias[col + 32]; bv3 = bias[col + 48]; }
#pragma unroll
    for (int r = 0; r < 8; ++r) {
        float* cr = C + (long)(row0 + r) * ldc + col;
        cr[0]  = acc0[r] + bv0;
        cr[16] = acc1[r] + bv1;
        cr[32] = acc2[r] + bv2;
        cr[48] = acc3[r] + bv3;
    }
}

// ---------------------------------------------------------------------------
// Row-wise softmax over [rows, cols], logits pre-scaled by `scale`.
// ---------------------------------------------------------------------------
__global__ __launch_bounds__(256) void
sru_softmax_rows(float* __restrict__ S, int cols, float scale)
{
    float* p = S + (long)blockIdx.x * cols;
    __shared__ float red[256];
    const int t = threadIdx.x;

    float mx = -3.402823466e38f;
    for (int c = t; c < cols; c += 256) mx = fmaxf(mx, p[c] * scale);
    red[t] = mx; __syncthreads();
    for (int s = 128; s > 0; s >>= 1) {
        if (t < s) red[t] = fmaxf(red[t], red[t + s]);
        __syncthreads();
    }
    mx = red[0]; __syncthreads();

    float sum = 0.f;
    for (int c = t; c < cols; c += 256) {
        float e = __expf(p[c] * scale - mx);
        p[c] = e;
        sum += e;
    }
    red[t] = sum; __syncthreads();
    for (int s = 128; s > 0; s >>= 1) {
        if (t < s) red[t] += red[t + s];
        __syncthreads();
    }
    const float inv = 1.f / red[0];
    for (int c = t; c < cols; c += 256) p[c] *= inv;
}

// Q += alpha * F (alpha read from device scalar)
__global__ __launch_bounds__(256) void
sru_axpy(float* __restrict__ Q, const float* __restrict__ F,
         const float* __restrict__ alpha, long n)
{
    const long i = (long)blockIdx.x * blockDim.x + threadIdx.x;
    if (i < n) Q[i] += alpha[0] * F[i];
}

// ---------------------------------------------------------------------------
// Sequential SRU recurrence. One thread per (b,h); coalesced over h.
// ---------------------------------------------------------------------------
__global__ __launch_bounds__(256) void
sru_scan(const float* __restrict__ U, const float* __restrict__ x,
         const float* __restrict__ vf, const float* __restrict__ vr,
         const float* __restrict__ bf, const float* __restrict__ br,
         float* __restrict__ Hs, int Bn, int Nn, int Hn)
{
    const int t = blockIdx.x * blockDim.x + threadIdx.x;
    if (t >= Bn * Hn) return;
    const int b = t / Hn, h = t % Hn;
    const float vfh = vf[h], vrh = vr[h], bfh = bf[h], brh = br[h];

    const long r3 = (long)3 * Hn;
    const float* ub = U + (long)b * Nn * r3 + h;
    const float* xb = x + (long)b * Nn * Hn + h;
    float*       hb = Hs + (long)b * Nn * Hn + h;

    float c = 0.f;
    for (int n = 0; n < Nn; ++n) {
        const float u0 = ub[0], u1 = ub[Hn], u2 = ub[2 * Hn];
        const float xt = xb[0];
        const float f  = 1.f / (1.f + __expf(-(u0 + vfh * c + bfh)));
        const float cn = f * c + (1.f - f) * u2;
        const float r  = 1.f / (1.f + __expf(-(u1 + vrh * c + brh)));
        hb[0] = r * cn + (1.f - r) * xt;
        c = cn;
        ub += r3; xb += Hn; hb += Hn;
    }
}

extern "C" void kernel_launch(void* const* d_in, const int* in_sizes, int n_in,
                              void* d_out, int out_size, void* d_ws, size_t ws_size,
                              hipStream_t stream)
{
    constexpr int B = 4, N = 2048, D = 1024, H = 1024;
    constexpr int BN = B * N;           // 8192
    constexpr int H3 = 3 * H;           // 3072

    const float* x    = (const float*)d_in[0];
    const float* W    = (const float*)d_in[1];
    const float* vf   = (const float*)d_in[2];
    const float* vr   = (const float*)d_in[3];
    const float* bf   = (const float*)d_in[4];
    const float* br   = (const float*)d_in[5];
    const float* Wq   = (const float*)d_in[6];
    const float* Wk   = (const float*)d_in[7];
    const float* Wv   = (const float*)d_in[8];
    const float* Wu_w = (const float*)d_in[9];
    const float* Wu_b = (const float*)d_in[10];
    const float* alpha= (const float*)d_in[11];
    float* out = (float*)d_out;

    // Workspace layout (floats). Overlaps chosen so peak = 48 Mfloat = 192 MB.
    float* ws = (float*)d_ws;
    const long Mf = 1l << 20;
    float* Q  = ws +  0 * Mf;   // [BN,H]   live: steps 1..8
    float* Kp = ws +  8 * Mf;   // [BN,H]   live: 2..4
    float* Vp = ws + 16 * Mf;   // [BN,H]   live: 3..6
    float* S  = ws + 24 * Mf;   // [B,N,N]  live: 4..6
    float* F  = ws + 40 * Mf;   // [BN,H]   live: 6..7
    float* U  = ws +  8 * Mf;   // [BN,3H]  live: 8..9 (reuses K/V/S-low)
    float* Hs = ws + 32 * Mf;   // [BN,H]   live: 9..10 (reuses S-high)

    const dim3 blk(256);
    auto nblk = [](int M, int Nc) { return (M / 128) * (Nc / 64); };

    // 1) Q = x @ Wq^T
    sru_gemm_wmma<true><<<dim3(nblk(BN, H), 1), blk, 0, stream>>>(
        x, D, 0, Wq, D, 0, Q, H, 0, nullptr, BN, H, D);
    // 2) K = Q @ Wk^T
    sru_gemm_wmma<true><<<dim3(nblk(BN, H), 1), blk, 0, stream>>>(
        Q, H, 0, Wk, H, 0, Kp, H, 0, nullptr, BN, H, H);
    // 3) V = Q @ Wv^T
    sru_gemm_wmma<true><<<dim3(nblk(BN, H), 1), blk, 0, stream>>>(
        Q, H, 0, Wv, H, 0, Vp, H, 0, nullptr, BN, H, H);
    // 4) S_b = Q_b @ K_b^T (batched over B)
    sru_gemm_wmma<true><<<dim3(nblk(N, N), B), blk, 0, stream>>>(
        Q, H, (long)N * H, Kp, H, (long)N * H, S, N, (long)N * N, nullptr, N, N, H);
    // 5) softmax rows with 1/sqrt(H) scale
    sru_softmax_rows<<<dim3(BN), blk, 0, stream>>>(S, N, 0.03125f);
    // 6) F_b = attn_b @ V_b
    sru_gemm_wmma<false><<<dim3(nblk(N, H), B), blk, 0, stream>>>(
        S, N, (long)N * N, Vp, H, (long)N * H, F, H, (long)N * H, nullptr, N, H, N);
    // 7) Q += alpha * F
    sru_axpy<<<dim3((BN * H) / 256), blk, 0, stream>>>(Q, F, alpha, (long)BN * H);
    // 8) U = QF @ Wu_w^T + Wu_b
    sru_gemm_wmma<true><<<dim3(nblk(BN, H3), 1), blk, 0, stream>>>(
        Q, H, 0, Wu_w, H, 0, U, H3, 0, Wu_b, BN, H3, H);
    // 9) recurrent scan -> Hs
    sru_scan<<<dim3((B * H) / 256), blk, 0, stream>>>(
        U, x, vf, vr, bf, br, Hs, B, N, H);
    // 10) out = Hs @ W^T
    sru_gemm_wmma<true><<<dim3(nblk(BN, H3), 1), blk, 0, stream>>>(
        Hs, H, 0, W, D, 0, out, H3, 0, nullptr, BN, H3, H);

    (void)in_sizes; (void)n_in; (void)out_size; (void)ws_size;
}